// OuterProductLinear_5471788335890
// MI455X (gfx1250) — compile-verified
//
#include <hip/hip_runtime.h>
#include <hip/hip_bf16.h>

typedef float v2f __attribute__((ext_vector_type(2)));
typedef float v4f __attribute__((ext_vector_type(4)));
typedef float v8f __attribute__((ext_vector_type(8)));

// ---------------------------------------------------------------------------
// Phase 1: t[n] = dot(x[n, :], v[:])   (x streamed once, 128 MiB)
// One wave32 per row; each lane strides 16B across K, then shfl-xor reduce.
// ---------------------------------------------------------------------------
__global__ __launch_bounds__(256) void opl_rank1_reduce(
    const float* __restrict__ x, const float* __restrict__ v,
    float* __restrict__ t, int K)
{
    const int lane = threadIdx.x & 31;
    const int row  = blockIdx.x * (blockDim.x >> 5) + (threadIdx.x >> 5);

    const v4f* xr = reinterpret_cast<const v4f*>(x + (size_t)row * K);
    const v4f* vv = reinterpret_cast<const v4f*>(v);

    float acc = 0.0f;
    const int nvec = K >> 2;                 // 1024 x 16B per row
    for (int i = lane; i < nvec; i += 32) {
        v4f a = __builtin_nontemporal_load(&xr[i]);  // x used once: NT
        v4f b = vv[i];                               // v reused: keep cached
        acc += a.x * b.x + a.y * b.y + a.z * b.z + a.w * b.w;
    }
    #pragma unroll
    for (int off = 16; off > 0; off >>= 1)
        acc += __shfl_xor(acc, off, 32);
    if (lane == 0) t[row] = acc;
}

// ---------------------------------------------------------------------------
// Phase 2: out[n, m] = t[n] * u[m] + bias[m]  via V_WMMA_F32_16X16X4_F32.
//   A (16x4 f32, 2 VGPRs): col K=0 = t[rows], K=1..3 zeroed.
//     Layout: VGPR0 lanes0-15 -> (M=lane, K=0); lanes16-31 -> K=2 (zeroed);
//             VGPR1 -> K=1/K=3 (zeroed).
//   B (4x16 f32, 2 VGPRs): row K=0 = u[cols] (lanes 0-15 of VGPR0), rest 0.
//   C/D (16x16 f32, 8 VGPRs): M = g + 8*(lane>=16), N = lane&15.
//     C initialized to bias[col] broadcast across all M.
// One wave owns a 16-row strip x 16 column-tiles (256 cols). EXEC stays all-1s
// (predication via selects only), as WMMA requires.
// ---------------------------------------------------------------------------
__global__ __launch_bounds__(256) void opl_rank1_expand_wmma(
    const float* __restrict__ t, const float* __restrict__ u,
    const float* __restrict__ bias, float* __restrict__ out,
    int N, int tilesPerWave)
{
    const int lane      = threadIdx.x & 31;
    const int gwave     = blockIdx.x * (blockDim.x >> 5) + (threadIdx.x >> 5);
    const int wavesPerStrip = (N / 16) / tilesPerWave;   // 16
    const int strip     = gwave / wavesPerStrip;
    const int chunk     = gwave % wavesPerStrip;
    const int row_base  = strip * 16;
    const int halfsel   = lane >> 4;     // 0 for lanes 0-15, 1 for lanes 16-31
    const int l15       = lane & 15;

    // A matrix: t down column K=0 (select, not branch -> EXEC stays full)
    const float tv = t[row_base + l15];
    v2f a;
    a.x = halfsel ? 0.0f : tv;           // VGPR0: K=0 (lanes0-15) / K=2 -> 0
    a.y = 0.0f;                          // VGPR1: K=1 / K=3

    const int col0 = chunk * tilesPerWave * 16;
    for (int jt = 0; jt < tilesPerWave; ++jt) {
        const int col_base = col0 + jt * 16;

        // B matrix: u across row K=0
        const float uv = u[col_base + l15];
        v2f b;
        b.x = halfsel ? 0.0f : uv;
        b.y = 0.0f;

        // C = bias broadcast down all 16 rows of the tile
        const float bv = bias[col_base + l15];
        v8f c;
        #pragma unroll
        for (int g = 0; g < 8; ++g) c[g] = bv;

        // D = A x B + C   (args: neg_a, A, neg_b, B, c_mod, C, reuse_a, reuse_b)
        v8f d = __builtin_amdgcn_wmma_f32_16x16x4_f32(
            false, a, false, b, (short)0, c, false, false);

        // Store per D layout: row = row_base + g + 8*halfsel, col = col_base+l15.
        // Each half-wave writes a contiguous 64B run per g -> coalesced.
        float* op = out + (size_t)(row_base + halfsel * 8) * N + col_base + l15;
        #pragma unroll
        for (int g = 0; g < 8; ++g)
            __builtin_nontemporal_store(d[g], &op[(size_t)g * N]);
    }
}

extern "C" void kernel_launch(void* const* d_in, const int* in_sizes, int n_in,
                              void* d_out, int out_size, void* d_ws, size_t ws_size,
                              hipStream_t stream) {
    (void)in_sizes; (void)n_in; (void)out_size; (void)ws_size;

    constexpr int N_TOKENS = 8192;
    constexpr int IN_F     = 4096;
    constexpr int OUT_F    = 4096;
    constexpr int TILES_PER_WAVE = 16;   // 256 output cols per wave

    const float* x    = (const float*)d_in[0];
    const float* u    = (const float*)d_in[1];   // [OUT_F, 1] flat
    const float* v    = (const float*)d_in[2];   // [1, IN_F] flat
    const float* bias = (const float*)d_in[3];
    float*       out  = (float*)d_out;
    float*       t    = (float*)d_ws;            // 8192 floats = 32 KiB scratch

    // Phase 1: 8192 waves (1 per row), 8 waves per 256-thread block.
    opl_rank1_reduce<<<N_TOKENS / 8, 256, 0, stream>>>(x, v, t, IN_F);

    // Phase 2: 512 strips * 16 waves/strip = 8192 waves -> 1024 blocks.
    const int strips = N_TOKENS / 16;
    const int wavesPerStrip = (OUT_F / 16) / TILES_PER_WAVE;
    const int totalWaves = strips * wavesPerStrip;
    opl_rank1_expand_wmma<<<totalWaves / 8, 256, 0, stream>>>(
        t, u, bias, out, OUT_F, TILES_PER_WAVE);
}